// Seg_lay_28200755265728
// MI455X (gfx1250) — compile-verified
//
#include <hip/hip_runtime.h>
#include <math.h>
#include <stdint.h>

typedef __attribute__((ext_vector_type(2))) float v2f;
typedef __attribute__((ext_vector_type(8))) float v8f;

#define N_Q    32768
#define N_P    8192
#define C_FROM 256
#define C_TO   128

// ---------------------------------------------------------------------------
// Async-copy helpers: GLOBAL_LOAD_ASYNC_TO_LDS_B128 (ASYNCcnt-tracked, no
// VGPR round-trip).  LDS destination = low 32 bits of the flat shared address
// (LDS aperture => addr[31:0] is the LDS byte offset).
// ---------------------------------------------------------------------------
__device__ __forceinline__
void async_b128_to_lds(float* lds_dst, const float* gsrc)
{
    const uint32_t lds = (uint32_t)(uintptr_t)lds_dst;
    asm volatile("global_load_async_to_lds_b128 %0, %1, off"
                 :: "v"(lds), "v"(gsrc)
                 : "memory");
}
__device__ __forceinline__ void wait_async_le1() {
    asm volatile("s_wait_asynccnt 0x1" ::: "memory");
}
__device__ __forceinline__ void wait_async_le0() {
    asm volatile("s_wait_asynccnt 0x0" ::: "memory");
}

// ---------------------------------------------------------------------------
// Kernel 0: pack source points as (px, py, pz, |p|^2) for the WMMA distance
// trick  d^2 = |p|^2 + |q|^2 - 2 p.q.
// ---------------------------------------------------------------------------
__global__ __launch_bounds__(256)
void pack_coords_kernel(const float* __restrict__ from_coords,
                        float* __restrict__ pack)          // [N_P*4]
{
    const int p = blockIdx.x * 256 + threadIdx.x;          // 8192 total
    const float x = from_coords[p * 3 + 0];
    const float y = from_coords[p * 3 + 1];
    const float z = from_coords[p * 3 + 2];
    pack[p * 4 + 0] = x;
    pack[p * 4 + 1] = y;
    pack[p * 4 + 2] = z;
    pack[p * 4 + 3] = x * x + y * y + z * z;
}

// Branchless top-3 insertion (ascending scores) with index tracking.
// Pure v_cmp + v_cndmask dataflow: no EXEC-mask divergence, pipelineable.
// Stage order matters: s2 reads old s1, s1 reads old s0.
__device__ __forceinline__
void ins3(float s, int p, float& s0, float& s1, float& s2,
          int& i0, int& i1, int& i2)
{
    const bool lt0 = s < s0;
    const bool lt1 = s < s1;
    const bool lt2 = s < s2;
    s2 = lt1 ? s1 : (lt2 ? s : s2);
    i2 = lt1 ? i1 : (lt2 ? p : i2);
    s1 = lt0 ? s0 : (lt1 ? s : s1);
    i1 = lt0 ? i0 : (lt1 ? p : i1);
    s0 = lt0 ? s : s0;
    i0 = lt0 ? p : i0;
}

// ---------------------------------------------------------------------------
// Kernel 1: WMMA-driven top-3 nearest neighbors.
// One wave owns 16 queries (N dim).  Per 16-candidate tile one
// V_WMMA_F32_16X16X4_F32 computes score = p.(-2q) + |p|^2 = d^2 - |q|^2 for
// 256 pairs (rank-identical to d^2 per query column).  Fragments:
//   A (16x4): lane(half,mn) = pack[cand_base+mn][2*half .. 2*half+1]  (ds b64)
//   B (4x16): lane(half,mn) = query mn components (2*half,2*half+1) of
//             (-2qx,-2qy,-2qz,1)  -- loaded ONCE per wave.
//   C (16x16): lane(half,mn) row r -> candidate 8*half+r, query mn.
// Each lane keeps top-3 over its half of the candidate set; halves merged at
// the end with __shfl_xor(.,16).  Candidates stream through 32 KB LDS chunks
// via async b128 copies.  d^2 = max(score + |q|^2, 0); weights = 1/d^2,
// normalized, NaN (coincident point) -> 1 as in the reference.
// ---------------------------------------------------------------------------
__global__ __launch_bounds__(256)
void knn_wmma_kernel(const float* __restrict__ pack,       // [N_P*4]
                     const float* __restrict__ to_coords,  // [N_Q*3]
                     int*   __restrict__ idx_out,          // [N_Q*3]
                     float* __restrict__ w_out)            // [N_Q*3]
{
    const int CH = 2048;                                   // cands per chunk
    __shared__ float sp[CH * 4];                           // 32 KB

    const int tid  = threadIdx.x;
    const int lane = tid & 31;
    const int half = lane >> 4;
    const int mn   = lane & 15;
    const int wave = tid >> 5;
    const int q    = blockIdx.x * 128 + wave * 16 + mn;    // this lane's query

    const float qx = to_coords[q * 3 + 0];
    const float qy = to_coords[q * 3 + 1];
    const float qz = to_coords[q * 3 + 2];
    const float qq = qx * qx + qy * qy + qz * qz;

    v2f b;                                                 // B fragment
    b.x = half ? (-2.f * qz) : (-2.f * qx);
    b.y = half ? 1.f         : (-2.f * qy);

    float s0 = INFINITY, s1 = INFINITY, s2 = INFINITY;
    int   i0 = 0, i1 = 0, i2 = 0;

    for (int base = 0; base < N_P; base += CH) {
        __syncthreads();                                   // sp free to refill
        #pragma unroll
        for (int t = 0; t < (CH * 4) / (256 * 4); ++t) {   // 8 x b128 / thread
            const int e = (tid + t * 256) * 4;             // float index
            async_b128_to_lds(sp + e, pack + (size_t)base * 4 + e);
        }
        wait_async_le0();                                  // own deposits done
        __syncthreads();                                   // all waves' done

        #pragma unroll 2
        for (int tb = 0; tb < CH; tb += 16) {
            const v2f a = *(const v2f*)(sp + (tb + mn) * 4 + 2 * half);
            v8f c = {0.f,0.f,0.f,0.f,0.f,0.f,0.f,0.f};
            c = __builtin_amdgcn_wmma_f32_16x16x4_f32(
                    false, a, false, b, (short)0, c, false, false);
            const int cbase = base + tb + 8 * half;
            #pragma unroll
            for (int r = 0; r < 8; ++r)
                ins3(c[r], cbase + r, s0, s1, s2, i0, i1, i2);
        }
    }

    // merge the two lane-halves' sorted top-3 triples
    const float o0 = __shfl_xor(s0, 16, 32);
    const float o1 = __shfl_xor(s1, 16, 32);
    const float o2 = __shfl_xor(s2, 16, 32);
    const int   j0 = __shfl_xor(i0, 16, 32);
    const int   j1 = __shfl_xor(i1, 16, 32);
    const int   j2 = __shfl_xor(i2, 16, 32);
    ins3(o0, j0, s0, s1, s2, i0, i1, i2);
    ins3(o1, j1, s0, s1, s2, i0, i1, i2);
    ins3(o2, j2, s0, s1, s2, i0, i1, i2);

    if (half == 0) {
        const float d0 = fmaxf(s0 + qq, 0.f);
        const float d1 = fmaxf(s1 + qq, 0.f);
        const float d2 = fmaxf(s2 + qq, 0.f);
        float w0 = 1.0f / d0, w1 = 1.0f / d1, w2 = 1.0f / d2;
        const float s = w0 + w1 + w2;
        w0 /= s; w1 /= s; w2 /= s;
        if (!(w0 == w0)) w0 = 1.0f;
        if (!(w1 == w1)) w1 = 1.0f;
        if (!(w2 == w2)) w2 = 1.0f;
        idx_out[q * 3 + 0] = i0; idx_out[q * 3 + 1] = i1; idx_out[q * 3 + 2] = i2;
        w_out[q * 3 + 0] = w0;   w_out[q * 3 + 1] = w1;   w_out[q * 3 + 2] = w2;
    }
}

// ---------------------------------------------------------------------------
// Kernel 2: X0[n] = concat(to_features[n], sum_k w[n,k]*from_features[idx[n,k]])
// One block per row, one thread per output channel (coalesced along C).
// ---------------------------------------------------------------------------
__global__ __launch_bounds__(384)
void gather_concat_kernel(const float* __restrict__ to_features,
                          const float* __restrict__ from_features,
                          const int*   __restrict__ idx,
                          const float* __restrict__ w,
                          float* __restrict__ X0)        // [N_Q, 384]
{
    const int row = blockIdx.x;
    const int c   = threadIdx.x;
    if (c < C_TO) {
        X0[(size_t)row * 384 + c] = to_features[(size_t)row * C_TO + c];
    } else {
        const int cc = c - C_TO;
        const int j0 = idx[row * 3 + 0], j1 = idx[row * 3 + 1], j2 = idx[row * 3 + 2];
        const float w0 = w[row * 3 + 0], w1 = w[row * 3 + 1], w2 = w[row * 3 + 2];
        const float acc = w0 * from_features[(size_t)j0 * C_FROM + cc]
                        + w1 * from_features[(size_t)j1 * C_FROM + cc]
                        + w2 * from_features[(size_t)j2 * C_FROM + cc];
        X0[(size_t)row * 384 + c] = acc;
    }
}

// ---------------------------------------------------------------------------
// Async copy of one 16x64 fp32 B-panel chunk (4 KB) global -> LDS.
// ---------------------------------------------------------------------------
__device__ __forceinline__
void async_copy_b_chunk(const float* __restrict__ W, int nout,
                        int k0, int n_base, float* BsBuf, int tid)
{
    const int e = tid * 4;            // element index within 16x64 chunk
    const int k = e >> 6;             // 0..15
    const int n = e & 63;             // multiple of 4 -> 16B aligned
    async_b128_to_lds(BsBuf + e, W + (size_t)(k0 + k) * nout + n_base + n);
}

// ---------------------------------------------------------------------------
// WMMA fp32 GEMM: Out = op(A) @ W + bias, op = identity or BN(scale,shift)+ReLU.
// 256 threads = 8 waves.  Block tile: 128 rows x 64 cols.  Wave w owns rows
// [blockIdx.y*128 + w*16) and ALL 64 cols (4 accumulators) -> each A fragment
// feeds 4 WMMAs.  The shared 16x64 B panel chunk is double-buffered in LDS via
// async global->LDS b128 copies; async loads complete in order, so
// s_wait_asynccnt<=1 releases the current chunk while the prefetch is in
// flight.  V_WMMA_F32_16X16X4_F32 fragment layout:
//   A (16x4):  lane half (lane>>4) picks K-pair {0,1}/{2,3}; row = lane&15.
//   B (4x16):  VGPR0 = rows K0(lanes0-15)/K2(lanes16-31), VGPR1 = K1/K3.
//   C/D:       VGPR r -> M = r + 8*(lane>>4), N = lane&15.
// ---------------------------------------------------------------------------
template <int KDIM, int NOUT, bool BN>
__global__ __launch_bounds__(256)
void wmma_gemm_kernel(const float* __restrict__ A,      // [N_Q, KDIM]
                      const float* __restrict__ W,      // [KDIM, NOUT]
                      const float* __restrict__ bias,   // [NOUT]
                      const float* __restrict__ scale,  // [KDIM] (BN only)
                      const float* __restrict__ shift,  // [KDIM] (BN only)
                      float* __restrict__ Out)          // [N_Q, NOUT]
{
    __shared__ float Bs[2][16 * 64];                    // 8 KB double buffer

    const int tid    = threadIdx.x;
    const int lane   = tid & 31;
    const int wave   = tid >> 5;
    const int half   = lane >> 4;
    const int mn     = lane & 15;
    const int m_base = blockIdx.y * 128 + wave * 16;
    const int n_base = blockIdx.x * 64;

    const float* arow = A + (size_t)(m_base + mn) * KDIM;

    v8f c0 = {0.f,0.f,0.f,0.f,0.f,0.f,0.f,0.f};
    v8f c1 = c0, c2 = c0, c3 = c0;

    const int NCHUNK = KDIM / 16;
    async_copy_b_chunk(W, NOUT, 0, n_base, Bs[0], tid);  // prologue prefetch

    for (int ch = 0; ch < NCHUNK; ++ch) {
        const int   cur = ch & 1;
        const int   k0  = ch * 16;
        const bool  more = (ch + 1) < NCHUNK;
        if (more)
            async_copy_b_chunk(W, NOUT, k0 + 16, n_base, Bs[cur ^ 1], tid);
        if (more) wait_async_le1(); else wait_async_le0();  // own deposits done
        __syncthreads();                                    // all waves' done
        const float* bsb = Bs[cur];

        #pragma unroll
        for (int kk = 0; kk < 4; ++kk) {
            const int kb = kk * 4 + 2 * half;
            v2f a = *(const v2f*)(arow + k0 + kb);
            if (BN) {
                const v2f sc = *(const v2f*)(scale + k0 + kb);
                const v2f sh = *(const v2f*)(shift + k0 + kb);
                float a0 = fmaf(a.x, sc.x, sh.x);
                float a1 = fmaf(a.y, sc.y, sh.y);
                a.x = a0 > 0.f ? a0 : 0.f;
                a.y = a1 > 0.f ? a1 : 0.f;
            }
            v2f b;
            b.x = bsb[(kb + 0) * 64 + 0 * 16 + mn];
            b.y = bsb[(kb + 1) * 64 + 0 * 16 + mn];
            c0 = __builtin_amdgcn_wmma_f32_16x16x4_f32(false, a, false, b, (short)0, c0, false, false);
            b.x = bsb[(kb + 0) * 64 + 1 * 16 + mn];
            b.y = bsb[(kb + 1) * 64 + 1 * 16 + mn];
            c1 = __builtin_amdgcn_wmma_f32_16x16x4_f32(false, a, false, b, (short)0, c1, false, false);
            b.x = bsb[(kb + 0) * 64 + 2 * 16 + mn];
            b.y = bsb[(kb + 1) * 64 + 2 * 16 + mn];
            c2 = __builtin_amdgcn_wmma_f32_16x16x4_f32(false, a, false, b, (short)0, c2, false, false);
            b.x = bsb[(kb + 0) * 64 + 3 * 16 + mn];
            b.y = bsb[(kb + 1) * 64 + 3 * 16 + mn];
            c3 = __builtin_amdgcn_wmma_f32_16x16x4_f32(false, a, false, b, (short)0, c3, false, false);
        }
        __syncthreads();          // finished reading Bs[cur] before next overwrite
    }

    float* orow = Out + (size_t)(m_base + 8 * half) * NOUT + n_base + mn;
    const float bc0 = bias[n_base +  0 + mn];
    const float bc1 = bias[n_base + 16 + mn];
    const float bc2 = bias[n_base + 32 + mn];
    const float bc3 = bias[n_base + 48 + mn];
    #pragma unroll
    for (int r = 0; r < 8; ++r) {
        orow[(size_t)r * NOUT +  0] = c0[r] + bc0;
        orow[(size_t)r * NOUT + 16] = c1[r] + bc1;
        orow[(size_t)r * NOUT + 32] = c2[r] + bc2;
        orow[(size_t)r * NOUT + 48] = c3[r] + bc3;
    }
}

// ---------------------------------------------------------------------------
// BatchNorm column statistics (deterministic two-stage tree; C fixed at 256).
// ---------------------------------------------------------------------------
__global__ __launch_bounds__(256)
void colstats_partial_kernel(const float* __restrict__ Y,  // [N_Q, 256]
                             float* __restrict__ ps,       // [128, 256]
                             float* __restrict__ pq)       // [128, 256]
{
    const int c   = threadIdx.x;
    const int blk = blockIdx.x;
    const int r0  = blk * (N_Q / 128);
    float s = 0.f, q = 0.f;
    for (int r = 0; r < N_Q / 128; ++r) {
        const float v = Y[(size_t)(r0 + r) * 256 + c];
        s += v;
        q += v * v;
    }
    ps[blk * 256 + c] = s;
    pq[blk * 256 + c] = q;
}

__global__ __launch_bounds__(256)
void colstats_final_kernel(const float* __restrict__ ps,
                           const float* __restrict__ pq,
                           const float* __restrict__ gamma,
                           const float* __restrict__ beta,
                           float* __restrict__ scale,
                           float* __restrict__ shift)
{
    const int c = threadIdx.x;
    float s = 0.f, q = 0.f;
    for (int b = 0; b < 128; ++b) {
        s += ps[b * 256 + c];
        q += pq[b * 256 + c];
    }
    const float mu   = s / (float)N_Q;
    const float var  = q / (float)N_Q - mu * mu;
    const float rstd = rsqrtf(var + 1e-5f);
    const float g    = gamma[c] * rstd;
    scale[c] = g;
    shift[c] = beta[c] - mu * g;
}

// ---------------------------------------------------------------------------
// Launch: pack -> knn(WMMA) -> gather/concat -> GEMM0(+bias) -> stats ->
//         GEMM1(BN+ReLU in) -> stats -> GEMM2(BN+ReLU in) -> d_out.
// ---------------------------------------------------------------------------
extern "C" void kernel_launch(void* const* d_in, const int* in_sizes, int n_in,
                              void* d_out, int out_size, void* d_ws, size_t ws_size,
                              hipStream_t stream) {
    const float* from_coords   = (const float*)d_in[0];
    const float* from_features = (const float*)d_in[1];
    const float* to_coords     = (const float*)d_in[2];
    const float* to_features   = (const float*)d_in[3];
    const float* W0 = (const float*)d_in[4];
    const float* b0 = (const float*)d_in[5];
    const float* W1 = (const float*)d_in[6];
    const float* b1 = (const float*)d_in[7];
    const float* W2 = (const float*)d_in[8];
    const float* b2 = (const float*)d_in[9];
    const float* gamma0 = (const float*)d_in[10];
    const float* beta0  = (const float*)d_in[11];
    const float* gamma1 = (const float*)d_in[12];
    const float* beta1  = (const float*)d_in[13];
    float* out = (float*)d_out;

    // Workspace carve-up (bytes, all well-aligned by construction)
    char* ws = (char*)d_ws;
    size_t off = 0;
    float* pck = (float*)(ws + off); off += (size_t)N_P * 4 * sizeof(float);  // 128 KB
    int*   idx = (int*)(ws + off);   off += (size_t)N_Q * 3 * sizeof(int);    // 384 KB
    float* wts = (float*)(ws + off); off += (size_t)N_Q * 3 * sizeof(float);  // 384 KB
    float* X0  = (float*)(ws + off); off += (size_t)N_Q * 384 * sizeof(float);// 48 MB
    float* Y0  = (float*)(ws + off); off += (size_t)N_Q * 256 * sizeof(float);// 32 MB
    float* Y1  = (float*)(ws + off); off += (size_t)N_Q * 256 * sizeof(float);// 32 MB
    float* ps  = (float*)(ws + off); off += (size_t)128 * 256 * sizeof(float);
    float* pq  = (float*)(ws + off); off += (size_t)128 * 256 * sizeof(float);
    float* scl = (float*)(ws + off); off += 256 * sizeof(float);
    float* shf = (float*)(ws + off); off += 256 * sizeof(float);
    (void)ws_size; (void)in_sizes; (void)n_in; (void)out_size;

    // 0) pack source coords as (x,y,z,|p|^2)
    pack_coords_kernel<<<N_P / 256, 256, 0, stream>>>(from_coords, pck);

    // 1) WMMA-driven KNN top-3 + weights
    knn_wmma_kernel<<<N_Q / 128, 256, 0, stream>>>(pck, to_coords, idx, wts);

    // 2) interpolation gather + concat -> X0 [N,384]
    gather_concat_kernel<<<N_Q, 384, 0, stream>>>(to_features, from_features, idx, wts, X0);

    // 3) layer 0: Y0 = X0 @ W0 + b0
    wmma_gemm_kernel<384, 256, false>
        <<<dim3(256 / 64, N_Q / 128), 256, 0, stream>>>(X0, W0, b0, nullptr, nullptr, Y0);

    // 4) BN0 statistics -> scale/shift
    colstats_partial_kernel<<<128, 256, 0, stream>>>(Y0, ps, pq);
    colstats_final_kernel<<<1, 256, 0, stream>>>(ps, pq, gamma0, beta0, scl, shf);

    // 5) layer 1: Y1 = relu(BN(Y0)) @ W1 + b1
    wmma_gemm_kernel<256, 256, true>
        <<<dim3(256 / 64, N_Q / 128), 256, 0, stream>>>(Y0, W1, b1, scl, shf, Y1);

    // 6) BN1 statistics
    colstats_partial_kernel<<<128, 256, 0, stream>>>(Y1, ps, pq);
    colstats_final_kernel<<<1, 256, 0, stream>>>(ps, pq, gamma1, beta1, scl, shf);

    // 7) layer 2: out = relu(BN(Y1)) @ W2 + b2   [N,128]
    wmma_gemm_kernel<256, 128, true>
        <<<dim3(128 / 64, N_Q / 128), 256, 0, stream>>>(Y1, W2, b2, scl, shf, out);
}